// GLIF_81166291959851
// MI455X (gfx1250) — compile-verified
//
#include <hip/hip_runtime.h>

// GLIF spiking-neuron scan, MI455X (gfx1250, wave32).
// T=64 sequential steps, elementwise over [B=128, N=2048].
// Memory-bound: ~134 MB HBM traffic => ~5.8us floor at 23.3 TB/s.
// No contraction dim anywhere => WMMA inapplicable; optimize the stream.

typedef float v4f __attribute__((ext_vector_type(4)));

#define TT 64
#define BB 128
#define NN 2048

__device__ __forceinline__ float sigf(float x) {
    return 1.0f / (1.0f + __expf(-x));
}

// ---------------------------------------------------------------------------
// Prep kernel 1: per-N fused constants into P[6*NN]:
//   P[0]=decay_v, P[1]=leak_term, P[2]=reset_w, P[3]=gamma_s, P[4]=Vth_s, P[5]=beta_s
// ---------------------------------------------------------------------------
__global__ void glif_params(const float* __restrict__ alpha, const float* __restrict__ beta,
                            const float* __restrict__ gamma, const float* __restrict__ tau,
                            const float* __restrict__ Vth,   const float* __restrict__ leak,
                            const float* __restrict__ reVth, float* __restrict__ P) {
    int n = blockIdx.x * blockDim.x + threadIdx.x;
    if (n >= NN) return;
    float al = sigf(alpha[n]);
    float be = sigf(beta[n]);
    float ga = sigf(gamma[n]);
    float ts = sigf(tau[n]);
    float vt = sigf(Vth[n]);
    float lk = sigf(leak[n]);
    float rv = sigf(reVth[n]);
    P[0 * NN + n] = 1.0f - al * (1.0f - ts);  // decay_v
    P[1 * NN + n] = (1.0f - al) * lk;         // leak_term
    P[2 * NN + n] = (1.0f - ga) * rv;         // reset_w
    P[3 * NN + n] = ga;
    P[4 * NN + n] = vt;
    P[5 * NN + n] = be;
}

// ---------------------------------------------------------------------------
// Prep kernel 2: W[t*NN+n] = 1 - beta_s[n] * (1 - sigmoid(conduct[t,n]))
// (512 KB [T,N] table; L2-resident, reused by all 128 batch rows)
// ---------------------------------------------------------------------------
__global__ void glif_w(const float* __restrict__ conduct, const float* __restrict__ P,
                       float* __restrict__ W) {
    int i = blockIdx.x * blockDim.x + threadIdx.x;
    if (i >= TT * NN) return;
    int n = i & (NN - 1);
    W[i] = 1.0f - P[5 * NN + n] * (1.0f - sigf(conduct[i]));
}

// ---------------------------------------------------------------------------
// Main streaming scan. One thread owns 4 consecutive neurons of one batch
// row; v/y state lives in registers across all T steps. 128-bit NT loads of
// tx, 128-bit NT stores of ty, L2-cached loads of W. Loop peeled so the
// steady-state body has unconditional next-step loads; unroll x4 so the
// scheduler can keep ~4 independent b128 loads in flight per wave (the grid
// is only ~2 waves/SIMD, so per-wave MLP is what hides HBM latency).
// ---------------------------------------------------------------------------
__global__ __launch_bounds__(256, 1)
void glif_scan(const float* __restrict__ tx, const float* __restrict__ P,
               const float* __restrict__ W, const float* __restrict__ conduct,
               int useW, float* __restrict__ out) {
    const int NQ  = NN / 4;
    int idx = blockIdx.x * 256 + threadIdx.x;      // 0 .. BB*NQ-1
    int b   = idx / NQ;                            // NQ = 512 (pow2 -> shift)
    int n0  = (idx - b * NQ) * 4;

    const v4f decay = *(const v4f*)(P + 0 * NN + n0);
    const v4f leakt = *(const v4f*)(P + 1 * NN + n0);
    const v4f resw  = *(const v4f*)(P + 2 * NN + n0);
    const v4f ga    = *(const v4f*)(P + 3 * NN + n0);
    const v4f vth   = *(const v4f*)(P + 4 * NN + n0);
    const v4f be    = *(const v4f*)(P + 5 * NN + n0);

    const size_t row = (size_t)BB * NN;            // tx/out stride per t (floats)
    const float* xp = tx  + (size_t)b * NN + n0;
    float*       op = out + (size_t)b * NN + n0;
    const float* wp = (useW ? W : conduct) + n0;   // stride NN per t

    v4f v = (v4f)0.0f;
    v4f y = (v4f)0.0f;

    // pipeline prologue: step-0 operands in registers
    v4f xc = __builtin_nontemporal_load((const v4f*)xp);
    v4f wc = *(const v4f*)wp;

#define GLIF_STEP(XC, WC)                                                     \
    do {                                                                      \
        v4f w_;                                                               \
        if (useW) {                                                           \
            w_ = (WC);                                                        \
        } else {                                                              \
            w_.x = 1.0f - be.x * (1.0f - sigf((WC).x));                       \
            w_.y = 1.0f - be.y * (1.0f - sigf((WC).y));                       \
            w_.z = 1.0f - be.z * (1.0f - sigf((WC).z));                       \
            w_.w = 1.0f - be.w * (1.0f - sigf((WC).w));                       \
        }                                                                     \
        v4f I_  = (XC) * w_;                                                  \
        v4f vv_ = decay * v;                                                  \
        vv_ = vv_ - ga * vv_ * y;          /* decay*v*(1 - ga*y) */           \
        vv_ = vv_ - leakt + I_ - resw * y;                                    \
        v4f yn_;                                                              \
        yn_.x = vv_.x > vth.x ? 1.0f : 0.0f;                                  \
        yn_.y = vv_.y > vth.y ? 1.0f : 0.0f;                                  \
        yn_.z = vv_.z > vth.z ? 1.0f : 0.0f;                                  \
        yn_.w = vv_.w > vth.w ? 1.0f : 0.0f;                                  \
        __builtin_nontemporal_store(yn_, (v4f*)op);                           \
        v = vv_; y = yn_;                                                     \
    } while (0)

    // steady state: unconditional next-step loads; unroll for in-flight MLP.
#pragma unroll 4
    for (int t = 0; t < TT - 1; ++t) {
        v4f xn = __builtin_nontemporal_load((const v4f*)(xp + row));
        v4f wn = *(const v4f*)(wp + NN);
        // distance-2 prefetch; speculative => tail overshoot silently dropped
        __builtin_prefetch(xp + 2 * row, 0, 0);

        GLIF_STEP(xc, wc);

        xc = xn; wc = wn;
        xp += row; op += row; wp += NN;
    }

    // epilogue: final step (t = TT-1), no further loads
    GLIF_STEP(xc, wc);

#undef GLIF_STEP
}

// ---------------------------------------------------------------------------
// Host-side launcher (graph-capture safe: only kernel launches on `stream`)
// Inputs (setup_inputs order): tx, alpha, beta, gamma, tau, Vth, leak, reVth, conduct
// ---------------------------------------------------------------------------
extern "C" void kernel_launch(void* const* d_in, const int* in_sizes, int n_in,
                              void* d_out, int out_size, void* d_ws, size_t ws_size,
                              hipStream_t stream) {
    const float* tx      = (const float*)d_in[0];
    const float* alpha   = (const float*)d_in[1];
    const float* beta    = (const float*)d_in[2];
    const float* gamma   = (const float*)d_in[3];
    const float* tau     = (const float*)d_in[4];
    const float* Vth     = (const float*)d_in[5];
    const float* leak    = (const float*)d_in[6];
    const float* reVth   = (const float*)d_in[7];
    const float* conduct = (const float*)d_in[8];
    float* out = (float*)d_out;

    float* P = (float*)d_ws;                  // 6*NN floats (48 KB)
    float* W = P + 6 * NN;                    // TT*NN floats (512 KB)
    const size_t needW = (size_t)(6 * NN + TT * NN) * sizeof(float);
    const int useW = (ws_size >= needW) ? 1 : 0;

    glif_params<<<(NN + 255) / 256, 256, 0, stream>>>(alpha, beta, gamma, tau, Vth,
                                                      leak, reVth, P);
    if (useW) {
        glif_w<<<(TT * NN + 255) / 256, 256, 0, stream>>>(conduct, P, W);
    }

    const int total = BB * NN / 4;            // 65536 threads, 2048 wave32s
    glif_scan<<<total / 256, 256, 0, stream>>>(tx, P, W, conduct, useW, out);
}